// GINWithSet2Set_39238821216993
// MI455X (gfx1250) — compile-verified
//
#include <hip/hip_runtime.h>
#include <hip/hip_bf16.h>

typedef __attribute__((ext_vector_type(2))) float v2f;
typedef __attribute__((ext_vector_type(8))) float v8f;

#define NNODES 50000
#define NEDGES 800000
#define NB     512
#define DIN    32
#define DHID   128

// ---------------------------------------------------------------------------
// utility
// ---------------------------------------------------------------------------
__global__ void fill_zero_kernel(float* __restrict__ p, long long n) {
    long long i = (long long)blockIdx.x * blockDim.x + threadIdx.x;
    if (i < n) p[i] = 0.0f;
}

// ---------------------------------------------------------------------------
// scatter-add:  agg[dst[e]*D + d] += x[src[e]*D + d]
// x/agg fit in L2 (25.6MB each) -> L2-resident atomics
// ---------------------------------------------------------------------------
__global__ __launch_bounds__(256)
void scatter_add_kernel(const float* __restrict__ x,
                        const int* __restrict__ src,
                        const int* __restrict__ dst,
                        float* __restrict__ agg,
                        long long total, int D) {
    long long idx = (long long)blockIdx.x * blockDim.x + threadIdx.x;
    if (idx >= total) return;
    int e = (int)(idx / D);
    int d = (int)(idx % D);
    int s = src[e];
    int t = dst[e];
    atomicAdd(&agg[(long long)t * D + d], x[(long long)s * D + d]);
}

// ---------------------------------------------------------------------------
// fp32 WMMA GEMM:  out[M x 128] = act( (A (+ Aadd)) [M x K] @ W [K x 128] + bias )
// Block = 256 threads = 8 waves; one 16-row M tile per block, wave w -> cols [16w,16w+16)
// A tile staged in LDS (padded stride K+4 -> conflict-free column reads).
// WMMA fp32 16x16x4 operand layout (ISA 7.12.2):
//   A: lanes 0-15: M=lane, {K0,K1}; lanes 16-31: M=lane-16, {K2,K3}
//   B: lanes 0-15: N=lane,  K0 in v0 / K1 in v1; lanes 16-31: K2/K3
//   C/D: vgpr r -> M = r (+8 for upper half-wave), N = lane%16
// ---------------------------------------------------------------------------
template<int K>
__global__ __launch_bounds__(256)
void gemm_bias_act_kernel(const float* __restrict__ A,
                          const float* __restrict__ Aadd,   // may be null
                          const float* __restrict__ W,      // K x 128 row-major
                          const float* __restrict__ bias,   // 128
                          float* __restrict__ out,          // M x 128
                          int relu) {
    __shared__ float Atile[16 * (K + 4)];
    const int tid  = threadIdx.x;
    const int wave = tid >> 5;        // 0..7 -> N tile
    const int lane = tid & 31;
    const int l16  = lane & 15;
    const int m0   = blockIdx.x * 16;
    const int n0   = wave * 16;

    // prefetch this wave's weight slab into cache
    #pragma unroll
    for (int kk = 0; kk < K; kk += 32)
        __builtin_prefetch(&W[kk * 128 + n0 + l16], 0, 0);

    // cooperative, coalesced load of the 16 x K A tile (+ optional residual add)
    for (int i = tid; i < 16 * K; i += 256) {
        int r = i / K, c = i % K;
        long long gi = (long long)(m0 + r) * K + c;
        float v = A[gi];
        if (Aadd) v += Aadd[gi];
        Atile[r * (K + 4) + c] = v;
    }
    __syncthreads();

    const int koff = (lane < 16) ? 0 : 2;   // upper half-wave holds K+2/K+3
    v8f acc = {0.f, 0.f, 0.f, 0.f, 0.f, 0.f, 0.f, 0.f};
    #pragma unroll
    for (int k0 = 0; k0 < K; k0 += 4) {
        int ka = k0 + koff;
        v2f a, b;
        a.x = Atile[l16 * (K + 4) + ka];
        a.y = Atile[l16 * (K + 4) + ka + 1];
        b.x = W[(long long)ka * 128 + n0 + l16];
        b.y = W[(long long)(ka + 1) * 128 + n0 + l16];
        acc = __builtin_amdgcn_wmma_f32_16x16x4_f32(
                  /*neg_a=*/false, a, /*neg_b=*/false, b,
                  /*c_mod=*/(short)0, acc, /*reuse_a=*/false, /*reuse_b=*/false);
    }

    float bcol = bias ? bias[n0 + l16] : 0.0f;
    #pragma unroll
    for (int r = 0; r < 8; ++r) {
        int row = m0 + r + ((lane < 16) ? 0 : 8);
        float v = acc[r] + bcol;
        if (relu) v = fmaxf(v, 0.0f);
        out[(long long)row * 128 + n0 + l16] = v;
    }
}

// ---------------------------------------------------------------------------
// Set2Set pieces
// ---------------------------------------------------------------------------
__global__ __launch_bounds__(256)
void lstm_gates_kernel(const float* __restrict__ qstar,   // B x 256
                       const float* __restrict__ hst,     // B x 128
                       const float* __restrict__ wih,     // 512 x 256
                       const float* __restrict__ whh,     // 512 x 128
                       const float* __restrict__ bih,
                       const float* __restrict__ bhh,
                       float* __restrict__ gates) {       // B x 512
    int idx = blockIdx.x * blockDim.x + threadIdx.x;
    if (idx >= NB * 512) return;
    int b = idx / 512, j = idx % 512;
    float s = bih[j] + bhh[j];
    const float* q  = qstar + (long long)b * 256;
    const float* wi = wih   + (long long)j * 256;
    for (int k = 0; k < 256; ++k) s += q[k] * wi[k];
    const float* h  = hst + (long long)b * 128;
    const float* wh = whh + (long long)j * 128;
    for (int k = 0; k < 128; ++k) s += h[k] * wh[k];
    gates[idx] = s;
}

__device__ __forceinline__ float sigmoidf_(float x) { return 1.0f / (1.0f + expf(-x)); }

__global__ __launch_bounds__(256)
void lstm_update_kernel(const float* __restrict__ gates,
                        float* __restrict__ cst, float* __restrict__ hst) {
    int idx = blockIdx.x * blockDim.x + threadIdx.x;
    if (idx >= NB * DHID) return;
    int b = idx / DHID, d = idx % DHID;
    const float* g = gates + (long long)b * 512;
    float ig = sigmoidf_(g[d]);
    float fg = sigmoidf_(g[128 + d]);
    float gg = tanhf(g[256 + d]);
    float og = sigmoidf_(g[384 + d]);
    float c  = fg * cst[idx] + ig * gg;
    cst[idx] = c;
    hst[idx] = og * tanhf(c);
}

// attention logits: e[n] = dot(x[n,:], hst[batch[n],:]) ; one wave32 per node
__global__ __launch_bounds__(256)
void attn_dot_kernel(const float* __restrict__ x, const float* __restrict__ hst,
                     const int* __restrict__ batch, float* __restrict__ eout) {
    int node = blockIdx.x * 8 + (threadIdx.x >> 5);
    int lane = threadIdx.x & 31;
    if (node >= NNODES) return;
    int b = batch[node];
    const float* xr = x   + (long long)node * DHID;
    const float* q  = hst + (long long)b    * DHID;
    float s = 0.0f;
    for (int i = lane; i < DHID; i += 32) s += xr[i] * q[i];
    for (int off = 16; off > 0; off >>= 1) s += __shfl_xor(s, off, 32);
    if (lane == 0) eout[node] = s;
}

// order-preserving float<->uint map for exact segment-max via atomicMax
__device__ __forceinline__ unsigned enc_f(float f) {
    unsigned u = __float_as_uint(f);
    return (u & 0x80000000u) ? ~u : (u | 0x80000000u);
}
__device__ __forceinline__ float dec_f(unsigned u) {
    return (u & 0x80000000u) ? __uint_as_float(u & 0x7FFFFFFFu) : __uint_as_float(~u);
}

__global__ __launch_bounds__(256)
void seg_max_kernel(const float* __restrict__ e, const int* __restrict__ batch,
                    unsigned* __restrict__ menc) {
    int n = blockIdx.x * blockDim.x + threadIdx.x;
    if (n >= NNODES) return;
    atomicMax(&menc[batch[n]], enc_f(e[n]));
}

__global__ __launch_bounds__(256)
void seg_exp_kernel(const float* __restrict__ e, const unsigned* __restrict__ menc,
                    const int* __restrict__ batch,
                    float* __restrict__ ex, float* __restrict__ denom) {
    int n = blockIdx.x * blockDim.x + threadIdx.x;
    if (n >= NNODES) return;
    int b = batch[n];
    float v = expf(e[n] - dec_f(menc[b]));
    ex[n] = v;
    atomicAdd(&denom[b], v);
}

__global__ __launch_bounds__(256)
void readout_kernel(const float* __restrict__ x, const float* __restrict__ ex,
                    const float* __restrict__ denom, const int* __restrict__ batch,
                    float* __restrict__ r) {
    long long idx = (long long)blockIdx.x * blockDim.x + threadIdx.x;
    if (idx >= (long long)NNODES * DHID) return;
    int n = (int)(idx / DHID), d = (int)(idx % DHID);
    int b = batch[n];
    float a = ex[n] / denom[b];
    atomicAdd(&r[(long long)b * DHID + d], a * x[idx]);
}

__global__ __launch_bounds__(256)
void build_qstar_kernel(const float* __restrict__ hst, const float* __restrict__ r,
                        float* __restrict__ qstar) {
    int idx = blockIdx.x * blockDim.x + threadIdx.x;
    if (idx >= NB * DHID) return;
    int b = idx / DHID, d = idx % DHID;
    qstar[(long long)b * 256 + d]       = hst[idx];
    qstar[(long long)b * 256 + 128 + d] = r[idx];
}

__global__ __launch_bounds__(256)
void head2_kernel(const float* __restrict__ h,      // B x 128
                  const float* __restrict__ fc2w,   // 128 x 3
                  const float* __restrict__ fc2b,   // 3
                  float* __restrict__ out) {        // B x 3
    int idx = blockIdx.x * blockDim.x + threadIdx.x;
    if (idx >= NB * 3) return;
    int b = idx / 3, o = idx % 3;
    float s = fc2b[o];
    const float* hr = h + (long long)b * 128;
    for (int d = 0; d < 128; ++d) s += hr[d] * fc2w[d * 3 + o];
    out[idx] = s;
}

// ---------------------------------------------------------------------------
// host launcher
// ---------------------------------------------------------------------------
static inline int blocks_for(long long n, int bs) { return (int)((n + bs - 1) / bs); }

extern "C" void kernel_launch(void* const* d_in, const int* in_sizes, int n_in,
                              void* d_out, int out_size, void* d_ws, size_t ws_size,
                              hipStream_t stream) {
    const float* x_in  = (const float*)d_in[0];          // N x 32
    const int*   ei    = (const int*)d_in[1];            // 2 x E
    const int*   batch = (const int*)d_in[2];            // N
    const int* src = ei;
    const int* dst = ei + NEDGES;

    const float* wa[4] = { (const float*)d_in[3],  (const float*)d_in[7],
                           (const float*)d_in[11], (const float*)d_in[15] };
    const float* ba[4] = { (const float*)d_in[4],  (const float*)d_in[8],
                           (const float*)d_in[12], (const float*)d_in[16] };
    const float* wb[4] = { (const float*)d_in[5],  (const float*)d_in[9],
                           (const float*)d_in[13], (const float*)d_in[17] };
    const float* bb[4] = { (const float*)d_in[6],  (const float*)d_in[10],
                           (const float*)d_in[14], (const float*)d_in[18] };
    const float* lstm_wih = (const float*)d_in[19];  // 512 x 256
    const float* lstm_whh = (const float*)d_in[20];  // 512 x 128
    const float* lstm_bih = (const float*)d_in[21];
    const float* lstm_bhh = (const float*)d_in[22];
    const float* fc1_w = (const float*)d_in[23];     // 256 x 128
    const float* fc1_b = (const float*)d_in[24];
    const float* fc2_w = (const float*)d_in[25];     // 128 x 3
    const float* fc2_b = (const float*)d_in[26];
    float* out = (float*)d_out;

    // workspace carve-up (floats)
    float* ws = (float*)d_ws;
    const long long ND = (long long)NNODES * DHID;
    float* X     = ws;               // N x 128 (layer output / current features)
    float* AGG   = X + ND;           // N x 128 (neighbor sums)
    float* H     = AGG + ND;         // N x 128 (MLP hidden)
    float* evec  = H + ND;           // N
    float* exv   = evec + NNODES;    // N
    float* hst   = exv + NNODES;     // B x 128
    float* cst   = hst + NB * DHID;  // B x 128
    float* qstar = cst + NB * DHID;  // B x 256
    float* gates = qstar + NB * 256; // B x 512
    float* menc  = gates + NB * 512; // B (unsigned, zero bits = -inf sentinel)
    float* denom = menc + NB;        // B
    float* rbuf  = denom + NB;       // B x 128
    float* headh = rbuf + NB * DHID; // B x 128

    const int MT = NNODES / 16;      // 3125 M tiles (exact)

    // ---- layer 0 (K=32 input) ----
    fill_zero_kernel<<<blocks_for((long long)NNODES * DIN, 256), 256, 0, stream>>>(AGG, (long long)NNODES * DIN);
    scatter_add_kernel<<<blocks_for((long long)NEDGES * DIN, 256), 256, 0, stream>>>(
        x_in, src, dst, AGG, (long long)NEDGES * DIN, DIN);
    gemm_bias_act_kernel<32><<<MT, 256, 0, stream>>>(AGG, x_in, wa[0], ba[0], H, 1);
    gemm_bias_act_kernel<128><<<MT, 256, 0, stream>>>(H, nullptr, wb[0], bb[0], X, 1);

    // ---- layers 1..3 ----
    for (int l = 1; l < 4; ++l) {
        fill_zero_kernel<<<blocks_for(ND, 256), 256, 0, stream>>>(AGG, ND);
        scatter_add_kernel<<<blocks_for((long long)NEDGES * DHID, 256), 256, 0, stream>>>(
            X, src, dst, AGG, (long long)NEDGES * DHID, DHID);
        gemm_bias_act_kernel<128><<<MT, 256, 0, stream>>>(AGG, X, wa[l], ba[l], H, 1);
        gemm_bias_act_kernel<128><<<MT, 256, 0, stream>>>(H, nullptr, wb[l], bb[l], X, 1);
    }

    // ---- Set2Set (3 steps) ----
    fill_zero_kernel<<<blocks_for(NB * DHID, 256), 256, 0, stream>>>(hst, NB * DHID);
    fill_zero_kernel<<<blocks_for(NB * DHID, 256), 256, 0, stream>>>(cst, NB * DHID);
    fill_zero_kernel<<<blocks_for(NB * 256, 256), 256, 0, stream>>>(qstar, NB * 256);

    for (int step = 0; step < 3; ++step) {
        lstm_gates_kernel<<<blocks_for(NB * 512, 256), 256, 0, stream>>>(
            qstar, hst, lstm_wih, lstm_whh, lstm_bih, lstm_bhh, gates);
        lstm_update_kernel<<<blocks_for(NB * DHID, 256), 256, 0, stream>>>(gates, cst, hst);
        attn_dot_kernel<<<blocks_for(NNODES, 8), 256, 0, stream>>>(X, hst, batch, evec);
        // reset per-step segment buffers (zero bits == -inf sentinel for menc)
        fill_zero_kernel<<<blocks_for(NB * (2 + DHID), 256), 256, 0, stream>>>(menc, NB * (2 + DHID));
        seg_max_kernel<<<blocks_for(NNODES, 256), 256, 0, stream>>>(evec, batch, (unsigned*)menc);
        seg_exp_kernel<<<blocks_for(NNODES, 256), 256, 0, stream>>>(evec, (const unsigned*)menc, batch, exv, denom);
        readout_kernel<<<blocks_for(ND, 256), 256, 0, stream>>>(X, exv, denom, batch, rbuf);
        build_qstar_kernel<<<blocks_for(NB * DHID, 256), 256, 0, stream>>>(hst, rbuf, qstar);
    }

    // ---- head ----
    gemm_bias_act_kernel<256><<<NB / 16, 256, 0, stream>>>(qstar, nullptr, fc1_w, fc1_b, headh, 1);
    head2_kernel<<<blocks_for(NB * 3, 256), 256, 0, stream>>>(headh, fc2_w, fc2_b, out);
}